// SparseMemory_35897336660606
// MI455X (gfx1250) — compile-verified
//
#include <hip/hip_runtime.h>
#include <math.h>

typedef __attribute__((ext_vector_type(2))) float v2f;
typedef __attribute__((ext_vector_type(4))) float v4f;
typedef __attribute__((ext_vector_type(8))) float v8f;

#define MEMN  65536
#define WDIM  128
#define KSEL  8
#define RNUM  9
#define BATCH 32
#define INDIM 512
#define IFACE 266   // 2*W + R + 1

// ---------------------------------------------------------------------------
// Kernel A: xi = x(32x512) @ W_if(512x266) + b_if, via V_WMMA_F32_16X16X4_F32.
// One wave32 per 16x16 output tile; K stepped by 4 (128 WMMA ops per wave).
// A 16x4 f32 layout (ISA 7.12.2): lanes 0-15 -> M=lane, regs {K0,K1};
//                                 lanes 16-31 -> M=lane-16, regs {K2,K3}.
// B 4x16 f32: reg0 = row K0 (lanes 0-15) / K2 (lanes 16-31); reg1 = K1 / K3.
// D: VGPR j -> M = j (lanes 0-15) or 8+j (lanes 16-31), N = lane & 15.
// Out-of-range columns (>=266): the B-column address is CLAMPED instead of
// predicated — D column j depends only on B column j, and those columns are
// never stored, so garbage accumulation there is harmless. This keeps the
// inner loop free of exec-mask branches (loads + v_wmma only).
// ---------------------------------------------------------------------------
__global__ __launch_bounds__(32)
void iface_gemm_wmma(const float* __restrict__ x, const float* __restrict__ Wif,
                     const float* __restrict__ bif, float* __restrict__ xi) {
    const int lane  = threadIdx.x;
    const int mBase = blockIdx.y * 16;
    const int nBase = blockIdx.x * 16;
    const int half  = (lane >= 16) ? 1 : 0;
    const int l16   = lane & 15;
    const int aRow  = mBase + l16;
    const int bCol  = nBase + l16;
    const int kOff  = half ? 2 : 0;
    const bool colOK = (bCol < IFACE);
    const int colC  = colOK ? bCol : (IFACE - 1);   // clamped, always in range

    v8f acc = {};
    for (int k = 0; k < INDIM; k += 4) {
        v2f a, b;
        const float* xp = x + aRow * INDIM + k + kOff;
        a.x = xp[0];
        a.y = xp[1];
        const float* wp = Wif + (k + kOff) * IFACE + colC;
        b.x = wp[0];
        b.y = wp[IFACE];
        acc = __builtin_amdgcn_wmma_f32_16x16x4_f32(
            /*neg_a=*/false, a, /*neg_b=*/false, b,
            /*c_mod=*/(short)0, acc, /*reuse_a=*/false, /*reuse_b=*/false);
    }

    if (colOK) {
        const float bias = bif[bCol];
#pragma unroll
        for (int j = 0; j < 8; ++j) {
            const int row = mBase + (half ? (8 + j) : j);
            xi[row * IFACE + bCol] = acc[j] + bias;
        }
    }
}

// ---------------------------------------------------------------------------
// Kernel B: gates + new_rv[b,r,w] = read_vectors + ww[b,r]*write_vector[b,w]
// ---------------------------------------------------------------------------
__global__ __launch_bounds__(256)
void build_newrv(const float* __restrict__ xi, const float* __restrict__ read_weights,
                 const float* __restrict__ read_vectors, float* __restrict__ new_rv) {
    const int idx = blockIdx.x * blockDim.x + threadIdx.x;
    if (idx >= BATCH * RNUM * WDIM) return;
    const int b = idx / (RNUM * WDIM);
    const int r = (idx / WDIM) % RNUM;
    const int w = idx % WDIM;
    const float* xb = xi + b * IFACE;
    const float g  = 1.0f / (1.0f + __expf(-xb[2 * WDIM + r]));   // gate
    const float wg = 1.0f / (1.0f + __expf(-xb[IFACE - 1]));      // write gate
    const float ww = wg * (g * read_weights[b * RNUM + r] + (1.0f - g));
    new_rv[idx] = read_vectors[idx] + ww * xb[WDIM + w];          // write_vector
}

// ---------------------------------------------------------------------------
// Kernel C: the 1-GiB HBM stream (99% of runtime; ~46 us floor @ 23.3 TB/s).
// d2[b,m] = ||row - q||^2 where the row is new_rv if m is one of the
// scattered positions (highest r wins). Non-temporal b128 loads keep the
// stream from evicting the 8 MB d2 array out of the 192 MB L2.
// ---------------------------------------------------------------------------
__global__ __launch_bounds__(256)
void dist_kernel(const float* __restrict__ memory, const float* __restrict__ xi,
                 const int* __restrict__ read_positions,
                 const float* __restrict__ new_rv, float* __restrict__ d2) {
    __shared__ float sq[WDIM];
    __shared__ int   spos[RNUM];
    const int b   = blockIdx.y;
    const int tid = threadIdx.x;
    if (tid < WDIM) sq[tid]   = xi[b * IFACE + tid];               // read_query
    if (tid < RNUM) spos[tid] = read_positions[b * RNUM + tid];
    __syncthreads();

    const int m = blockIdx.x * 256 + tid;
    int src_r = -1;
#pragma unroll
    for (int r = RNUM - 1; r >= 0; --r)
        if (src_r < 0 && spos[r] == m) src_r = r;

    const float* row = (src_r >= 0)
        ? (new_rv + (b * RNUM + src_r) * WDIM)
        : (memory + ((size_t)b * MEMN + m) * (size_t)WDIM);

    float acc = 0.0f;
#pragma unroll
    for (int w = 0; w < WDIM; w += 4) {
        const v4f v = __builtin_nontemporal_load((const v4f*)(row + w));
        const float d0  = v.x - sq[w + 0];
        const float d1  = v.y - sq[w + 1];
        const float d2c = v.z - sq[w + 2];
        const float d3  = v.w - sq[w + 3];
        acc += d0 * d0 + d1 * d1 + d2c * d2c + d3 * d3;
    }
    d2[(size_t)b * MEMN + m] = acc;
}

// ---------------------------------------------------------------------------
// Kernel D: per-batch top-8 smallest d2. Register bubble-insert per thread
// over a contiguous 256-element chunk (stable, earliest-index on ties),
// then LDS merge by thread 0 with an explicit index tiebreak.
// ---------------------------------------------------------------------------
__global__ __launch_bounds__(256)
void topk_kernel(const float* __restrict__ d2, int* __restrict__ topk) {
    __shared__ float sv[256 * KSEL];
    __shared__ int   si[256 * KSEL];
    const int b   = blockIdx.x;
    const int tid = threadIdx.x;

    float vals[KSEL]; int idxs[KSEL];
#pragma unroll
    for (int j = 0; j < KSEL; ++j) { vals[j] = 3.4e38f; idxs[j] = 0x7fffffff; }

    const int chunk = MEMN / 256;
    const int base  = tid * chunk;
    const float* dd = d2 + (size_t)b * MEMN;
    for (int i = 0; i < chunk; ++i) {
        float cv = dd[base + i]; int ci = base + i;
#pragma unroll
        for (int j = 0; j < KSEL; ++j) {
            if (cv < vals[j]) {
                float tv = vals[j]; int ti = idxs[j];
                vals[j] = cv; idxs[j] = ci; cv = tv; ci = ti;
            }
        }
    }
#pragma unroll
    for (int j = 0; j < KSEL; ++j) { sv[tid * KSEL + j] = vals[j]; si[tid * KSEL + j] = idxs[j]; }
    __syncthreads();

    if (tid == 0) {
        float bv[KSEL]; int bi[KSEL];
#pragma unroll
        for (int j = 0; j < KSEL; ++j) { bv[j] = 3.4e38f; bi[j] = 0x7fffffff; }
        for (int t = 0; t < 256 * KSEL; ++t) {
            float cv = sv[t]; int ci = si[t];
#pragma unroll
            for (int j = 0; j < KSEL; ++j) {
                if (cv < bv[j] || (cv == bv[j] && ci < bi[j])) {
                    float tv = bv[j]; int ti = bi[j];
                    bv[j] = cv; bi[j] = ci; cv = tv; ci = ti;
                }
            }
        }
#pragma unroll
        for (int j = 0; j < KSEL; ++j) topk[b * KSEL + j] = bi[j];
    }
}

// ---------------------------------------------------------------------------
// Kernel E: gather the K selected rows (updated view). last_used_mem row is
// dropped by the reference's [:, :-1] so it never reaches the output.
// ---------------------------------------------------------------------------
__global__ __launch_bounds__(128)
void gather_kernel(const float* __restrict__ memory, const float* __restrict__ new_rv,
                   const int* __restrict__ read_positions, const int* __restrict__ topk,
                   float* __restrict__ out) {
    const int b = blockIdx.y;
    const int k = blockIdx.x;
    const int w = threadIdx.x;
    int pos = topk[b * KSEL + k];
    if (pos < 0) pos = 0;
    if (pos >= MEMN) pos = MEMN - 1;
    int src_r = -1;
#pragma unroll
    for (int r = RNUM - 1; r >= 0; --r)
        if (src_r < 0 && read_positions[b * RNUM + r] == pos) src_r = r;
    const float* row = (src_r >= 0)
        ? (new_rv + (b * RNUM + src_r) * WDIM)
        : (memory + ((size_t)b * MEMN + pos) * (size_t)WDIM);
    out[((b * KSEL) + k) * WDIM + w] = row[w];
}

extern "C" void kernel_launch(void* const* d_in, const int* in_sizes, int n_in,
                              void* d_out, int out_size, void* d_ws, size_t ws_size,
                              hipStream_t stream) {
    (void)in_sizes; (void)n_in; (void)out_size; (void)ws_size;
    const float* x            = (const float*)d_in[0];
    const float* Wif          = (const float*)d_in[1];
    const float* bif          = (const float*)d_in[2];
    const float* memory       = (const float*)d_in[3];
    const float* read_weights = (const float*)d_in[4];
    const float* read_vectors = (const float*)d_in[5];
    const int*   read_pos     = (const int*)d_in[6];
    // d_in[7] = last_used_mem: dead (dropped by reference's [:, :-1]).

    float* ws     = (float*)d_ws;
    float* xi     = ws;                                 // 32*266      =   8512 f
    float* new_rv = xi + BATCH * IFACE;                 // 32*9*128    =  36864 f
    float* d2     = new_rv + BATCH * RNUM * WDIM;       // 32*65536    = 2097152 f
    int*   topk   = (int*)(d2 + (size_t)BATCH * MEMN);  // 32*8 ints

    dim3 gA((IFACE + 15) / 16, BATCH / 16);             // 17 x 2 tiles, 1 wave each
    iface_gemm_wmma<<<gA, 32, 0, stream>>>(x, Wif, bif, xi);

    const int nB = BATCH * RNUM * WDIM;
    build_newrv<<<(nB + 255) / 256, 256, 0, stream>>>(xi, read_weights, read_vectors, new_rv);

    dim3 gC(MEMN / 256, BATCH);                         // 8192 blocks: stream 1 GiB
    dist_kernel<<<gC, 256, 0, stream>>>(memory, xi, read_pos, new_rv, d2);

    topk_kernel<<<BATCH, 256, 0, stream>>>(d2, topk);

    dim3 gE(KSEL, BATCH);
    gather_kernel<<<gE, 128, 0, stream>>>(memory, new_rv, read_pos, topk, (float*)d_out);
}